// FourierAttn1_5970004542053
// MI455X (gfx1250) — compile-verified
//
#include <hip/hip_runtime.h>

// ---------------------------------------------------------------------------
// Problem constants (from the reference)
// ---------------------------------------------------------------------------
#define QLEN   1024
#define MLEN   1024
#define KLEN   (QLEN + MLEN)   // 2048
#define BSZ    4
#define NHEAD  8
#define DHEAD  16
#define DMODEL 128
#define EPS_F  1e-6f

// score-kernel tiling
#define TQ 16        // q rows per block (8 waves x 2 q each)
#define TK 8         // k rows per block (LDS tables: 48 KB)

typedef float v2f __attribute__((ext_vector_type(2)));
typedef float v8f __attribute__((ext_vector_type(8)));

// ---------------------------------------------------------------------------
// Kernel 1: f32 projection GEMM via V_WMMA_F32_16X16X4_F32.
// Computes out[row, col] = sum_k A[row, k] * W[k, col], A is [rows,128],
// W is [128,128].  One 16x16 output tile per wave, 32 chained K=4 WMMAs.
// The concat(mems, h) of the reference is handled with a per-tile pointer
// select: rows < splitRow come from A0, rows >= splitRow from A1.
// ---------------------------------------------------------------------------
__global__ __launch_bounds__(256)
void proj_wmma(const float* __restrict__ A0, const float* __restrict__ A1,
               int splitRow, const float* __restrict__ W,
               float* __restrict__ out, int rowTiles) {
  const int wave = (blockIdx.x * blockDim.x + threadIdx.x) >> 5;
  const int lane = threadIdx.x & 31;
  const int colTile = wave & 7;          // 128 / 16 = 8 column tiles
  const int rowTile = wave >> 3;
  if (rowTile >= rowTiles) return;       // whole wave exits together (EXEC all-1 otherwise)

  const int rowBase = rowTile * 16;
  const float* __restrict__ A =
      (rowBase < splitRow) ? (A0 + (size_t)rowBase * DMODEL)
                           : (A1 + (size_t)(rowBase - splitRow) * DMODEL);

  // ISA 7.12.2, 32-bit A 16x4: lanes 0-15 hold K=+0,+1 ; lanes 16-31 K=+2,+3
  const int m     = lane & 15;
  const int kHalf = (lane >> 4) << 1;    // 0 or 2
  const int colBase = colTile * 16;

  v8f c = {0.f, 0.f, 0.f, 0.f, 0.f, 0.f, 0.f, 0.f};
#pragma unroll
  for (int kk = 0; kk < DMODEL / 4; ++kk) {
    const int k0 = kk * 4 + kHalf;
    v2f a, b;
    a.x = A[m * DMODEL + k0];
    a.y = A[m * DMODEL + k0 + 1];
    b.x = W[(k0)     * DMODEL + colBase + m];
    b.y = W[(k0 + 1) * DMODEL + colBase + m];
    // 8 args: (neg_a, A, neg_b, B, c_mod, C, reuse_a, reuse_b)
    c = __builtin_amdgcn_wmma_f32_16x16x4_f32(false, a, false, b,
                                              (short)0, c, false, false);
  }

  // C layout: VGPR i -> row rowBase + i (lanes 0-15) / rowBase + 8 + i (16-31)
  const int n = colBase + m;
  const int rowOff = rowBase + ((lane >> 4) << 3);
#pragma unroll
  for (int i = 0; i < 8; ++i)
    out[(size_t)(rowOff + i) * DMODEL + n] = c[i];
}

// ---------------------------------------------------------------------------
// Kernel 2: Fourier-integral (sinc-product) scores.
//   out[q,k,b,h] = | prod_d ( |diff|<eps ? R_h : sin(R_h*diff)/diff ) |,
//   diff = hq[q,b,h,d] - hk[k,b,h,d].
// sin(R*(q-k)) = sin(Rq)cos(Rk) - cos(Rq)sin(Rk): per-element sincos is
// precomputed (k tiles in LDS, q tiles in registers), so the inner loop is
// FMA-only with 2 reciprocals per (q,k) pair (two groups of 8 factors).
// Lane -> (h = lane&7, b = lane>>3): a wave writes 128 contiguous bytes.
// ---------------------------------------------------------------------------
__global__ __launch_bounds__(256)
void fourier_scores(const float* __restrict__ hq, const float* __restrict__ hk,
                    const float* __restrict__ paramR, float* __restrict__ out) {
  __shared__ float  s_kv [TK * DHEAD * 32];   // 16 KB  k values
  __shared__ float2 s_ksc[TK * DHEAD * 32];   // 32 KB  (sin, cos) of R*k

  const int t    = threadIdx.x;
  const int lane = t & 31;
  const int wave = t >> 5;
  const int h    = lane & 7;
  const int b    = lane >> 3;
  const int qBase = blockIdx.x * TQ;
  const int kBase = blockIdx.y * TK;
  const float R   = paramR[h];

  // ---- build k-tile tables in LDS: TK*16*32 = 4096 elements, 16/thread ----
#pragma unroll
  for (int it = 0; it < (TK * DHEAD * 32) / 256; ++it) {
    const int idx = it * 256 + t;          // == (kl*DHEAD + d)*32 + l
    const int kl  = idx >> 9;
    const int rem = idx & 511;
    const int d   = rem >> 5;
    const int l   = rem & 31;
    const int hh  = l & 7, bb = l >> 3;
    const float v = hk[((size_t)(kBase + kl) * BSZ + bb) * DMODEL + hh * DHEAD + d];
    float sn, cs;
    __sincosf(paramR[hh] * v, &sn, &cs);
    s_kv[idx]  = v;
    s_ksc[idx] = make_float2(sn, cs);
  }

  // ---- build this wave's 2 q tables in registers (96 VGPRs) ----
  float qv[2][DHEAD], qs[2][DHEAD], qc[2][DHEAD];
  const int q0 = qBase + wave * 2;
#pragma unroll
  for (int qi = 0; qi < 2; ++qi) {
    const float* __restrict__ p =
        &hq[((size_t)(q0 + qi) * BSZ + b) * DMODEL + h * DHEAD];
#pragma unroll
    for (int d = 0; d < DHEAD; ++d) {
      const float v = p[d];
      qv[qi][d] = v;
      __sincosf(R * v, &qs[qi][d], &qc[qi][d]);
    }
  }
  __syncthreads();

  // ---- main loop: for each k in tile, product over d for both q's ----
  for (int k = 0; k < TK; ++k) {
    float prod0 = 1.f, prod1 = 1.f;
    float num0 = 1.f, den0 = 1.f, num1 = 1.f, den1 = 1.f;
#pragma unroll
    for (int d = 0; d < DHEAD; ++d) {
      const int idx = (k * DHEAD + d) * 32 + lane;
      const float  kv = s_kv[idx];
      const float2 sc = s_ksc[idx];
      {
        const float diff  = qv[0][d] - kv;
        const float sn    = qs[0][d] * sc.y - qc[0][d] * sc.x;
        const bool  small = fabsf(diff) < EPS_F;
        num0 *= small ? R   : sn;
        den0 *= small ? 1.f : diff;
      }
      {
        const float diff  = qv[1][d] - kv;
        const float sn    = qs[1][d] * sc.y - qc[1][d] * sc.x;
        const bool  small = fabsf(diff) < EPS_F;
        num1 *= small ? R   : sn;
        den1 *= small ? 1.f : diff;
      }
      if (d == 7) {  // fold first group of 8 to bound under/overflow
        prod0 = num0 * __builtin_amdgcn_rcpf(den0); num0 = 1.f; den0 = 1.f;
        prod1 = num1 * __builtin_amdgcn_rcpf(den1); num1 = 1.f; den1 = 1.f;
      }
    }
    prod0 *= num0 * __builtin_amdgcn_rcpf(den0);
    prod1 *= num1 * __builtin_amdgcn_rcpf(den1);

    const int kk = kBase + k;
    out[((size_t)q0       * KLEN + kk) * 32 + lane] = fabsf(prod0);
    out[((size_t)(q0 + 1) * KLEN + kk) * 32 + lane] = fabsf(prod1);
  }
}

// ---------------------------------------------------------------------------
// Host launcher.  Inputs: h [1024,4,128], mems [1024,4,128], Wq [128,128],
// Wk [128,128], paramR [8].  Output: [1024,2048,4,8] f32.
// Workspace: head_q (2 MB) + head_k (4 MB) at the front of d_ws.
// ---------------------------------------------------------------------------
extern "C" void kernel_launch(void* const* d_in, const int* in_sizes, int n_in,
                              void* d_out, int out_size, void* d_ws, size_t ws_size,
                              hipStream_t stream) {
  const float* h    = (const float*)d_in[0];
  const float* mems = (const float*)d_in[1];
  const float* Wq   = (const float*)d_in[2];
  const float* Wk   = (const float*)d_in[3];
  const float* R    = (const float*)d_in[4];
  float* out = (float*)d_out;

  float* head_q = (float*)d_ws;                          // [4096, 128]
  float* head_k = head_q + (size_t)QLEN * BSZ * DMODEL;  // [8192, 128]

  // head_q = h @ Wq : 4096/16 = 256 row tiles * 8 col tiles = 2048 waves
  proj_wmma<<<256, 256, 0, stream>>>(h, h, QLEN * BSZ, Wq, head_q, 256);
  // head_k = concat(mems, h) @ Wk : 8192/16 = 512 row tiles -> 4096 waves
  proj_wmma<<<512, 256, 0, stream>>>(mems, h, MLEN * BSZ, Wk, head_k, 512);
  // scores
  fourier_scores<<<dim3(QLEN / TQ, KLEN / TK), 256, 0, stream>>>(
      head_q, head_k, R, out);
}